// QuantumMemoryNetwork_6657199309113
// MI455X (gfx1250) — compile-verified
//
#include <hip/hip_runtime.h>

typedef __attribute__((ext_vector_type(2))) float v2f;
typedef __attribute__((ext_vector_type(8))) float v8f;

#define Bq 32
#define Sq 512
#define Dq 64
#define Tq 16            // chunk length == WMMA M (time rows)
#define NCHUNK (Sq / Tq) // 32
#define DECAY 0.9f

// ---------------- Phase 1a: per-chunk weighted outer-product sums ----------------
// sums[b][c][i][j] = sum_{k=0..15} decay^(15-k) * c_k[i]*c_k[j]   (chunk-local)
__global__ void __launch_bounds__(256)
qmn_chunk_sums(const int* __restrict__ x, const float* __restrict__ emb,
               float* __restrict__ sums) {
  const int blk = blockIdx.x;  // b*NCHUNK + c
  const int b = blk >> 5;
  const int c = blk & 31;
  const int tid = threadIdx.x;

  __shared__ float cont[Tq][Dq];  // 4 KB
  for (int q = tid; q < Tq * Dq; q += 256) {
    int t = q >> 6, d = q & 63;
    int tok = x[b * Sq + c * Tq + t];
    cont[t][d] = emb[tok * Dq + d];
  }
  __syncthreads();

  float* dst = sums + (size_t)blk * (Dq * Dq);
  for (int e = tid; e < Dq * Dq; e += 256) {
    int i = e >> 6, j = e & 63;
    float acc = 0.f, w = 1.f;
#pragma unroll
    for (int k = Tq - 1; k >= 0; --k) {  // w = decay^(15-k)
      acc += w * cont[k][i] * cont[k][j];
      w *= DECAY;
    }
    dst[e] = acc;
  }
}

// ---------------- Phase 1b: scan chunk sums -> chunk-entry boundary states -------
// In place: buf[b][c] becomes rho entering chunk c (rho before chunk 0 = I).
__global__ void __launch_bounds__(256)
qmn_scan_boundaries(float* __restrict__ buf) {
  const int b = blockIdx.x;
  const int tid = threadIdx.x;

  float d16 = 1.f;
#pragma unroll
  for (int q = 0; q < Tq; ++q) d16 *= DECAY;  // decay^16

  float rho[16];
#pragma unroll
  for (int q = 0; q < 16; ++q) {
    int e = tid + q * 256;
    rho[q] = ((e >> 6) == (e & 63)) ? 1.f : 0.f;  // identity
  }
  for (int c = 0; c < NCHUNK; ++c) {
    float* p = buf + ((size_t)(b * NCHUNK + c)) * (Dq * Dq);
#pragma unroll
    for (int q = 0; q < 16; ++q) {
      int e = tid + q * 256;
      float s = p[e];
      p[e] = rho[q];               // boundary entering chunk c
      rho[q] = d16 * rho[q] + s;   // advance past chunk c
    }
  }
}

// ---------------- Phase 2: WMMA decay-prefix matmul + streaming stores -----------
// out[t, ij] = decay^(t+1)*rho_prev[ij] + sum_k M[t,k] * (c_k[i]*c_k[j]),
// M[t,k] = decay^(t-k) for k<=t (lower triangular). One 16x16 output tile per
// 5x V_WMMA_F32_16X16X4_F32 (1 rho_prev injection + 4 K-slices).
__global__ void __launch_bounds__(256)
qmn_chunk_states(const int* __restrict__ x, const float* __restrict__ emb,
                 const float* __restrict__ bnd, float* __restrict__ out) {
  const int blk = blockIdx.x;  // b*NCHUNK + c
  const int b = blk >> 5;
  const int c = blk & 31;
  const int tid = threadIdx.x;
  const int lane = tid & 31;
  const int wave = tid >> 5;   // 8 waves/block
  const int half = lane >> 4;  // lane halves (K / row split per ISA layout)
  const int lp = lane & 15;

  __shared__ float cont[Tq][Dq];    // 4 KB: gathered chunk content
  __shared__ float rhoP[Dq * Dq];   // 16 KB: rho entering this chunk
  __shared__ float dpow[Tq + 1];    // decay^0 .. decay^16

  if (tid < Tq + 1) {
    float p = 1.f;
    for (int q = 0; q < tid; ++q) p *= DECAY;
    dpow[tid] = p;
  }
  for (int q = tid; q < Tq * Dq; q += 256) {
    int t = q >> 6, d = q & 63;
    int tok = x[b * Sq + c * Tq + t];
    cont[t][d] = emb[tok * Dq + d];
  }
  const float* bsrc = bnd + (size_t)blk * (Dq * Dq);
  for (int q = tid; q < Dq * Dq; q += 256) rhoP[q] = bsrc[q];
  __syncthreads();

  // A-matrix lane layout (32-bit A 16x4): lanes 0-15 hold K={4s,4s+1} for row
  // m=lane; lanes 16-31 hold K={4s+2,4s+3} for row m=lane-16.
  const int m = lp;  // time row within chunk
  v2f A[4];
#pragma unroll
  for (int s = 0; s < 4; ++s) {
    int k0 = 4 * s + 2 * half;
    A[s].x = (k0 <= m) ? dpow[m - k0] : 0.f;
    A[s].y = (k0 + 1 <= m) ? dpow[m - k0 - 1] : 0.f;
  }
  v2f Aprev;                                   // injects decay^(t+1)*rho_prev
  Aprev.x = (half == 0) ? dpow[m + 1] : 0.f;   // only K=0 column is live
  Aprev.y = 0.f;

  // Branchless per-lane mask: B row K=0 lives in vgpr0 of lanes 0-15 only.
  const float rhoMask = (half == 0) ? 1.f : 0.f;

  float* obase = out + ((size_t)(b * Sq + c * Tq)) * (Dq * Dq);

  // 4096 rho columns -> 256 tiles of 16; 8 waves round-robin (uniform per wave,
  // EXEC stays all-ones around WMMA; no divergent branches in the loop body).
  for (int tile = wave; tile < (Dq * Dq) / 16; tile += 8) {
    int i = tile >> 2;          // rho row index
    int j0 = (tile & 3) * 16;   // rho col base (constant per wave)

    // B-matrix 4x16: VGPR v, lane L -> row K = v + 2*(L>=16), col N = L&15.
    v2f Bp;
    Bp.x = rhoMask * rhoP[i * Dq + j0 + lp];  // unconditional LDS read
    Bp.y = 0.f;
    v8f acc = {};
    acc = __builtin_amdgcn_wmma_f32_16x16x4_f32(false, Aprev, false, Bp,
                                                (short)0, acc, false, false);
#pragma unroll
    for (int s = 0; s < 4; ++s) {
      int k0 = 4 * s + 2 * half;
      v2f Bk;
      Bk.x = cont[k0][i] * cont[k0][j0 + lp];
      Bk.y = cont[k0 + 1][i] * cont[k0 + 1][j0 + lp];
      acc = __builtin_amdgcn_wmma_f32_16x16x4_f32(false, A[s], false, Bk,
                                                  (short)0, acc, false, false);
    }
    // C/D 16x16 layout: VGPR r, lane L -> row m = r + 8*(L>=16), col = L&15.
    // Output is written exactly once and never re-read: non-temporal stores
    // keep the 256 MB stream from evicting emb / boundary data in L2.
#pragma unroll
    for (int r = 0; r < 8; ++r) {
      int mt = r + 8 * half;
      __builtin_nontemporal_store(acc[r],
                                  &obase[(size_t)mt * (Dq * Dq) + i * Dq + j0 + lp]);
    }
  }
}

extern "C" void kernel_launch(void* const* d_in, const int* in_sizes, int n_in,
                              void* d_out, int out_size, void* d_ws, size_t ws_size,
                              hipStream_t stream) {
  (void)in_sizes; (void)n_in; (void)out_size; (void)ws_size;
  const int* x = (const int*)d_in[0];          // [32, 512] token ids
  const float* emb = (const float*)d_in[1];    // [32000, 64] f32
  float* out = (float*)d_out;                  // [32, 512, 64, 64] f32
  float* ws = (float*)d_ws;  // 32*32*4096 f32 = 16 MB: chunk sums -> boundaries

  qmn_chunk_sums<<<dim3(Bq * NCHUNK), dim3(256), 0, stream>>>(x, emb, ws);
  qmn_scan_boundaries<<<dim3(Bq), dim3(256), 0, stream>>>(ws);
  qmn_chunk_states<<<dim3(Bq * NCHUNK), dim3(256), 0, stream>>>(x, emb, ws, out);
}